// CRFLayer_4827543240880
// MI455X (gfx1250) — compile-verified
//
#include <hip/hip_runtime.h>

// CDNA5 / gfx1250. wave32. Matrix core: V_WMMA_F32_16X16X4_F32 (fp32 in/out,
// matches reference precision). Problem is HBM-bound (AI ~16 F/B), so we
// compute only the mask-compacted rows (~halves HBM traffic) and keep W
// (128 KB) L2-resident.

typedef float v2f __attribute__((ext_vector_type(2)));
typedef float v8f __attribute__((ext_vector_type(8)));

static constexpr int kB = 16;    // batch
static constexpr int kS = 2048;  // subtokens
static constexpr int kD = 1024;  // hidden
static constexpr int kL = 32;    // labels

// ---------------------------------------------------------------- rowlist init
__global__ void init_rowlist_kernel(int* __restrict__ rowlist, int n) {
    int i = blockIdx.x * blockDim.x + threadIdx.x;
    if (i < n) rowlist[i] = -1;
}

// ------------------------------------------------------- mask scan / compaction
// One wave (32 lanes) per batch row. Ballot-based exclusive scan over S=2048.
__global__ void mask_scan_kernel(const int* __restrict__ mask,
                                 int* __restrict__ rowlist,
                                 int* __restrict__ lens, int max_tok) {
    const int b    = blockIdx.x;
    const int lane = threadIdx.x;  // 0..31
    int base = 0;
    for (int c = 0; c < kS; c += 32) {
        const int m = mask[b * kS + c + lane];
        const unsigned bal = (unsigned)__ballot(m != 0);           // wave32: low 32 bits
        const int pre = __popc(bal & ((1u << lane) - 1u));         // exclusive prefix
        if (m) rowlist[b * max_tok + base + pre] = b * kS + c + lane;
        base += __popc(bal);
    }
    if (lane == 0) lens[b] = base;
}

// ------------------------------------------------------------------- WMMA GEMM
// One wave per 16-row output tile. N=32 as two 16-wide accumulators.
// K handled in segments (blockIdx.y) of Kseg, accumulated into partial buffers.
//
// A fragment (32-bit 16x4, ISA 7.12.2): lane = M (mod 16); lanes 0-15 hold
// K={0,1}, lanes 16-31 hold K={2,3} -> one float2 load per lane per step.
// B fragment mirrors with lane = N.
__global__ __launch_bounds__(256)
void gemm_tiles_kernel(const float* __restrict__ emb,
                       const float* __restrict__ W,
                       const int* __restrict__ rowlist,
                       float* __restrict__ partial,  // [kspl][R*kL]
                       int numTiles, int R, int Kseg) {
    const int wave = (blockIdx.x * blockDim.x + threadIdx.x) >> 5;  // wave-uniform
    if (wave >= numTiles) return;                                   // uniform exit
    const int lane = threadIdx.x & 31;
    const int half = lane >> 4;   // 0: K={0,1}, 1: K={2,3}
    const int lm   = lane & 15;   // M for A, N for B
    const int kk   = half * 2;
    const int k0   = blockIdx.y * Kseg;

    const int tileBase = wave * 16;
    int src = rowlist[tileBase + lm];
    if (src < 0) src = 0;  // pad row: compute garbage, finalize() discards it

    const float* aRow  = emb + (long)src * kD + k0 + kk;
    const float* bRow0 = W + (long)lm * kD + k0 + kk;
    const float* bRow1 = W + (long)(lm + 16) * kD + k0 + kk;

    v8f acc0 = {0.f, 0.f, 0.f, 0.f, 0.f, 0.f, 0.f, 0.f};
    v8f acc1 = {0.f, 0.f, 0.f, 0.f, 0.f, 0.f, 0.f, 0.f};

#pragma unroll 4
    for (int k = 0; k < Kseg; k += 4) {
        v2f a  = *(const v2f*)(aRow + k);
        v2f b0 = *(const v2f*)(bRow0 + k);
        v2f b1 = *(const v2f*)(bRow1 + k);
        // (neg_a, A, neg_b, B, c_mod, C, reuse_a, reuse_b)
        acc0 = __builtin_amdgcn_wmma_f32_16x16x4_f32(false, a, false, b0,
                                                     (short)0, acc0, false, false);
        acc1 = __builtin_amdgcn_wmma_f32_16x16x4_f32(false, a, false, b1,
                                                     (short)0, acc1, false, false);
    }

    // D layout: VGPR r, lane -> M = r + 8*half, N = lane&15.
    float* p = partial + (long)blockIdx.y * (long)R * kL;
#pragma unroll
    for (int r = 0; r < 8; ++r) {
        const int outRow = tileBase + r + half * 8;
        if (outRow < R) {
            p[(long)outRow * kL + lm]      = acc0[r];
            p[(long)outRow * kL + 16 + lm] = acc1[r];
        }
    }
}

// -------------------------------------------------------------------- finalize
// out[0 .. R*L)      : sum of K partials + bias, zeroed for pad rows
// out[R*L .. R*L+R)  : pad_mask as float (t < token_lens[b])
__global__ void finalize_kernel(const float* __restrict__ partial,
                                const float* __restrict__ bias,
                                const int* __restrict__ lens,
                                float* __restrict__ out,
                                int R, int max_tok, int kspl, long total) {
    const long RL = (long)R * kL;
    for (long i = (long)blockIdx.x * blockDim.x + threadIdx.x; i < total;
         i += (long)gridDim.x * blockDim.x) {
        if (i < RL) {
            const int outRow = (int)(i / kL);
            const int n      = (int)(i % kL);
            const int b      = outRow / max_tok;
            const int t      = outRow % max_tok;
            float v = 0.f;
            if (t < lens[b]) {
                v = bias[n];
                for (int s = 0; s < kspl; ++s) v += partial[(long)s * RL + i];
            }
            out[i] = v;
        } else {
            const long j = i - RL;
            const int b  = (int)(j / max_tok);
            const int t  = (int)(j % max_tok);
            out[i] = (t < lens[b]) ? 1.0f : 0.0f;
        }
    }
}

// ---------------------------------------------------------------------- launch
extern "C" void kernel_launch(void* const* d_in, const int* in_sizes, int n_in,
                              void* d_out, int out_size, void* d_ws, size_t ws_size,
                              hipStream_t stream) {
    const float* emb  = (const float*)d_in[0];  // [B,S,D] f32
    const int*   mask = (const int*)d_in[1];    // [B,S] i32
    const float* W    = (const float*)d_in[2];  // [L,D] f32
    const float* bias = (const float*)d_in[3];  // [L] f32
    float* out = (float*)d_out;

    // out = tok_logits [B,max_tok,L] ++ pad_mask [B,max_tok]  => 16*33*max_tok
    int max_tok = out_size / (kB * (kL + 1));
    if (max_tok < 1) max_tok = 1;
    const int  R        = kB * max_tok;
    const int  numTiles = (R + 15) / 16;
    const long RL       = (long)R * kL;

    // workspace layout: rowlist | lens | partials
    char*  ws      = (char*)d_ws;
    int*   rowlist = (int*)ws;
    size_t off     = (size_t)numTiles * 16 * sizeof(int);
    int*   lens    = (int*)(ws + off);
    off += 16 * sizeof(int);
    off = (off + 255) & ~(size_t)255;
    float* partial = (float*)(ws + off);
    const size_t avail = (ws_size > off) ? (ws_size - off) : 0;

    // K-split for occupancy (~1100 tiles alone under-fills the GPU); pick the
    // largest split whose partial buffers fit in the workspace. Deterministic.
    int kspl = 0;
    if ((size_t)8 * RL * sizeof(float) <= avail)      kspl = 8;
    else if ((size_t)4 * RL * sizeof(float) <= avail) kspl = 4;
    else if ((size_t)2 * RL * sizeof(float) <= avail) kspl = 2;
    else if ((size_t)1 * RL * sizeof(float) <= avail) kspl = 1;
    if (kspl == 0) {  // ws too small: accumulate in place in d_out (kspl=1)
        kspl    = 1;
        partial = out;  // finalize reads/writes same element -> race-free
    }
    const int Kseg = kD / kspl;

    init_rowlist_kernel<<<(numTiles * 16 + 255) / 256, 256, 0, stream>>>(
        rowlist, numTiles * 16);
    mask_scan_kernel<<<kB, 32, 0, stream>>>(mask, rowlist, lens, max_tok);

    dim3 grid((numTiles + 7) / 8, kspl);  // 8 waves (tiles) per 256-thread block
    gemm_tiles_kernel<<<grid, 256, 0, stream>>>(emb, W, rowlist, partial,
                                                numTiles, R, Kseg);

    const long total  = RL + R;
    int blocks = (int)((total + 255) / 256);
    if (blocks > 4096) blocks = 4096;
    finalize_kernel<<<blocks, 256, 0, stream>>>(partial, bias, lens, out,
                                                R, max_tok, kspl, total);
}